// SpikeSoftmax_36558761624012
// MI455X (gfx1250) — compile-verified
//
#include <hip/hip_runtime.h>
#include <stdint.h>

// SpikeSoftmax on MI455X (gfx1250).
// Bandwidth-bound: ~201 MB traffic, ~8.6 us floor at 23.3 TB/s. No GEMM
// structure -> WMMA inapplicable; use the CDNA5 async global->LDS DMA path
// (ASYNCcnt) for prefetch + wave32 shuffle reductions + B128 accesses.
// Streaming data (read-once / write-once) -> non-temporal hints to avoid
// churning the 192 MB L2.

typedef float v4f __attribute__((ext_vector_type(4)));

#define TT 4
#define BB 2
#define HH 12
#define LL 512
#define DD 512
#define ROWS (BB * HH * LL)                 // 12288 rows, one wave each
#define PLANE ((size_t)BB * HH * LL * DD)   // elements between t-planes
#define WAVES_PER_BLOCK 8
#define THREADS (WAVES_PER_BLOCK * 32)

__launch_bounds__(THREADS)
__global__ void spike_softmax_kernel(const float* __restrict__ x,
                                     float* __restrict__ out) {
    // 8 waves * 4 planes * 512 floats = 64 KB LDS
    __shared__ float buf[WAVES_PER_BLOCK * TT * DD];

    const int wave = threadIdx.x >> 5;
    const int lane = threadIdx.x & 31;
    const int r = blockIdx.x * WAVES_PER_BLOCK + wave;   // row id (b,h,l)

    const int b  = r / (HH * LL);
    const int hl = r % (HH * LL);
    const size_t base0 = (size_t)b * ((size_t)HH * LL * DD) + (size_t)hl * DD;

    float* mybuf = &buf[(wave * TT) * DD];

    // ---- issue async prefetch of all 4 t-planes of this row (ASYNCcnt += 16)
    //      non-temporal: streamed once, keep it out of L2's way
#pragma unroll
    for (int t = 0; t < TT; ++t) {
        const float* gsrc = x + base0 + (size_t)t * PLANE;
        const uint32_t lbase = (uint32_t)(uintptr_t)(&mybuf[t * DD]);
#pragma unroll
        for (int k = 0; k < 4; ++k) {
            uint64_t ga = (uint64_t)(uintptr_t)(gsrc + (size_t)(lane + 32 * k) * 4);
            uint32_t la = lbase + (uint32_t)((lane + 32 * k) * 16);
            asm volatile("global_load_async_to_lds_b128 %0, %1, off th:TH_LOAD_NT"
                         :: "v"(la), "v"(ga)
                         : "memory");
        }
    }

    // ---- scan carries live entirely in registers (single pass over memory)
    float Xsum[16], Phi[16], corr[16];
#pragma unroll
    for (int i = 0; i < 16; ++i) { Xsum[i] = 0.f; Phi[i] = 0.f; corr[i] = 0.f; }
    float Ysum = 0.f;
    const float inv_tp1[TT] = {1.0f, 0.5f, 1.0f / 3.0f, 0.25f};

#pragma unroll
    for (int t = 0; t < TT; ++t) {
        // wait until this plane's 4 async loads have landed (loads complete
        // in order; allow the remaining (3-t)*4 to stay in flight)
        if      (t == 0) asm volatile("s_wait_asynccnt 12" ::: "memory");
        else if (t == 1) asm volatile("s_wait_asynccnt 8"  ::: "memory");
        else if (t == 2) asm volatile("s_wait_asynccnt 4"  ::: "memory");
        else             asm volatile("s_wait_asynccnt 0"  ::: "memory");

        // read 16 row elements from LDS as 128-bit vectors (ds_load_b128)
        float v[16];
        const v4f* lp = (const v4f*)(&mybuf[t * DD]);
#pragma unroll
        for (int k = 0; k < 4; ++k) {
            v4f q = lp[lane + 32 * k];
            v[4 * k + 0] = q.x; v[4 * k + 1] = q.y;
            v[4 * k + 2] = q.z; v[4 * k + 3] = q.w;
        }

        // row max: 16-elem local reduce + 5 wave32 xor-shuffles
        float m = v[0];
#pragma unroll
        for (int i = 1; i < 16; ++i) m = fmaxf(m, v[i]);
#pragma unroll
        for (int off = 16; off > 0; off >>= 1)
            m = fmaxf(m, __shfl_xor(m, off, 32));

        // exps and row sum
        float e[16];
        float s = 0.f;
#pragma unroll
        for (int i = 0; i < 16; ++i) { e[i] = __expf(v[i] - m); s += e[i]; }
#pragma unroll
        for (int off = 16; off > 0; off >>= 1)
            s += __shfl_xor(s, off, 32);
        const float Yt = 1.0f / s;

        // scan update + output
        float o[16];
#pragma unroll
        for (int i = 0; i < 16; ++i) {
            Xsum[i] += e[i];
            const float phi  = Xsum[i] * Yt + e[i] * Ysum;
            Phi[i] += phi;
            const float tpsi = Phi[i] * inv_tp1[t];
            o[i] = tpsi - corr[i];
            corr[i] = tpsi;
        }
        Ysum += Yt;

        // coalesced non-temporal B128 stores (output never re-read)
        v4f* op = (v4f*)(out + base0 + (size_t)t * PLANE);
#pragma unroll
        for (int k = 0; k < 4; ++k) {
            v4f q;
            q.x = o[4 * k + 0]; q.y = o[4 * k + 1];
            q.z = o[4 * k + 2]; q.w = o[4 * k + 3];
            __builtin_nontemporal_store(q, &op[lane + 32 * k]);
        }
    }
}

extern "C" void kernel_launch(void* const* d_in, const int* in_sizes, int n_in,
                              void* d_out, int out_size, void* d_ws, size_t ws_size,
                              hipStream_t stream) {
    const float* x = (const float*)d_in[0];
    float* out = (float*)d_out;
    dim3 grid(ROWS / WAVES_PER_BLOCK);   // 1536 blocks
    dim3 block(THREADS);                 // 256 threads = 8 wave32
    hipLaunchKernelGGL(spike_softmax_kernel, grid, block, 0, stream, x, out);
}